// LastAggregator_48447231098887
// MI455X (gfx1250) — compile-verified
//
#include <hip/hip_runtime.h>

#ifndef __has_builtin
#define __has_builtin(x) 0
#endif

// CDNA5 async global<->LDS copy path (gfx1250): guarded so we compile either way.
#if __has_builtin(__builtin_amdgcn_global_load_async_to_lds_b128) && \
    __has_builtin(__builtin_amdgcn_global_store_async_from_lds_b128)
#define LA_ASYNC 1
#else
#define LA_ASYNC 0
#endif

typedef unsigned long long u64;
// Native clang vectors (HIP's float4 is a class type; builtins want raw vectors).
typedef float v4f __attribute__((vector_size(16)));
typedef int   v4i __attribute__((vector_size(16)));

#define GLOBAL_AS __attribute__((address_space(1)))
#define LDS_AS    __attribute__((address_space(3)))

#if LA_ASYNC
__device__ __forceinline__ GLOBAL_AS v4i* la_to_global_v4i(const void* p) {
  // two-step: pointee reinterpret in generic space, then addrspace cast
  return (GLOBAL_AS v4i*)(v4i*)p;
}
__device__ __forceinline__ LDS_AS v4i* la_to_lds_v4i(void* p) {
  return (LDS_AS v4i*)(v4i*)p;
}
#endif

__device__ __forceinline__ void la_wait_async0() {
#if __has_builtin(__builtin_amdgcn_s_wait_asynccnt)
  __builtin_amdgcn_s_wait_asynccnt(0);
#else
  asm volatile("s_wait_asynccnt 0x0" ::: "memory");
#endif
}

// ---------------------------------------------------------------------------
// Kernel 1: zero the per-node key table (harness poisons d_ws with 0xAA).
// ---------------------------------------------------------------------------
__global__ __launch_bounds__(256) void la_init_keys(u64* __restrict__ keys, int n) {
  int i = blockIdx.x * 256 + threadIdx.x;
  if (i < n) keys[i] = 0ull;
}

// ---------------------------------------------------------------------------
// Kernel 2: scatter-argmax. Pack (t, pos) into one u64 so a single
// global_atomic_max_u64 implements "max t, ties -> largest pos".
//   key = (order_preserving_bits(t) << 32) | (pos + 1);  key==0 <=> empty node
// ---------------------------------------------------------------------------
__global__ __launch_bounds__(256) void la_scatter_argmax(const float* __restrict__ t,
                                                         const int* __restrict__ index,
                                                         u64* __restrict__ keys,
                                                         int E) {
  int e = blockIdx.x * 256 + threadIdx.x;
  if (e >= E) return;
  unsigned tb = __float_as_uint(t[e]);
  // monotone float->uint map (handles negatives too; uniform[0,1) is the easy case)
  tb = (tb & 0x80000000u) ? ~tb : (tb | 0x80000000u);
  u64 key = ((u64)tb << 32) | (u64)(unsigned)(e + 1);
  atomicMax(&keys[index[e]], key);  // -> global_atomic_max_u64, resolves in L2
}

// ---------------------------------------------------------------------------
// Kernel 3: masked row gather. One wave32 per node: 32 lanes x 16B = 512B
// = one full D=128 row per wave, b128 per lane. Winning rows are copied
// global->LDS->global with the CDNA5 async copy engine (ASYNCcnt), bypassing
// VGPRs; empty nodes stream zeros with nontemporal b128 stores.
// ---------------------------------------------------------------------------
__global__ __launch_bounds__(256) void la_gather(const float* __restrict__ msg,
                                                 const u64* __restrict__ keys,
                                                 float* __restrict__ out,
                                                 int N, int D) {
#if LA_ASYNC
  __shared__ float slots[8 * 128];  // 8 waves/block * 512B staging
#endif
  const int lane = threadIdx.x & 31;
  const int wv   = threadIdx.x >> 5;
  const int node = blockIdx.x * 8 + wv;
  if (node >= N) return;

  const u64 key = keys[node];  // wave-uniform (same node for all 32 lanes)
  float* dst = out + (size_t)node * D;

  if (key == 0ull) {  // node never received a message -> zeros
    const v4f z = {0.f, 0.f, 0.f, 0.f};
    for (int d = lane * 4; d < D; d += 128)
      __builtin_nontemporal_store(z, (v4f*)(dst + d));
    return;
  }

  const unsigned pos = (unsigned)(key & 0xffffffffull) - 1u;
  const float* src = msg + (size_t)pos * D;

#if LA_ASYNC
  float* slot = &slots[wv * 128 + lane * 4];
  for (int d = lane * 4; d < D; d += 128) {
    __builtin_amdgcn_global_load_async_to_lds_b128(
        la_to_global_v4i(src + d), la_to_lds_v4i(slot),
        /*imm offset*/ 0, /*cpol*/ 0);
    la_wait_async0();  // LDS slot filled
    __builtin_amdgcn_global_store_async_from_lds_b128(
        la_to_global_v4i(dst + d), la_to_lds_v4i(slot),
        /*imm offset*/ 0, /*cpol*/ 0);
    la_wait_async0();  // slot reusable next chunk
  }
#else
  for (int d = lane * 4; d < D; d += 128) {
    v4f v = __builtin_nontemporal_load((const v4f*)(src + d));
    __builtin_nontemporal_store(v, (v4f*)(dst + d));
  }
#endif
}

// ---------------------------------------------------------------------------
// Inputs (setup_inputs order): msg [E*D f32], index [E i32], t [E f32],
// dim_size [1 i32]. Output: [N*D f32]. d_ws: N u64 keys (800 KB).
// ---------------------------------------------------------------------------
extern "C" void kernel_launch(void* const* d_in, const int* in_sizes, int n_in,
                              void* d_out, int out_size, void* d_ws, size_t ws_size,
                              hipStream_t stream) {
  (void)n_in; (void)ws_size;
  const float* msg   = (const float*)d_in[0];
  const int*   index = (const int*)d_in[1];
  const float* t     = (const float*)d_in[2];

  const int E = in_sizes[1];
  const int D = (E > 0) ? (in_sizes[0] / E) : 128;   // 128
  const int N = (D > 0) ? (out_size / D) : 0;        // 100000

  u64* keys = (u64*)d_ws;

  la_init_keys     <<<(N + 255) / 256, 256, 0, stream>>>(keys, N);
  la_scatter_argmax<<<(E + 255) / 256, 256, 0, stream>>>(t, index, keys, E);
  la_gather        <<<(N + 7) / 8,     256, 0, stream>>>(msg, keys, (float*)d_out, N, D);
}